// CorrelationModule_3607772528921
// MI455X (gfx1250) — compile-verified
//
#include <hip/hip_runtime.h>
#include <hip/hip_bf16.h>

typedef float v2f __attribute__((ext_vector_type(2)));
typedef float v8f __attribute__((ext_vector_type(8)));

#define H_    64
#define W_    128
#define P_    8192              // H*W
#define ND_   81
#define MTOT_ 663552            // 81*8192
#define EPS_  1e-5f

// ---------------- workspace layout (float offsets) ----------------
#define OFF_STATS1 0u                       // 192  (sum|sumsq, 96 ch)
#define OFF_STATS2 192u                     // 256  (128 ch)
#define OFF_STATS3 448u                     // 128  (64 ch)
#define OFF_AB1    576u                     // 192
#define OFF_AB2    768u                     // 256
#define OFF_AB3    1024u                    // 128
#define OFF_T1     1280u                    // 8192*96  : t1[p][o]  (W1a . f1)
#define OFF_G      (OFF_T1 + 8192u*96u)     // 8192*96  : g[p][o]   (W1b . f2)
#define OFF_Y1     (OFF_G  + 8192u*96u)     // 663552*96  (also reused as y3: 663552*64)
#define OFF_Y2     (OFF_Y1 + (size_t)MTOT_*96u) // 663552*128

// ------------------------------------------------------------------
__global__ __launch_bounds__(256) void init_zero(float* p, int n) {
  int i = blockIdx.x * 256 + threadIdx.x;
  if (i < n) p[i] = 0.0f;
}

// ------------------------------------------------------------------
// Generic fp32 WMMA GEMM:  Y[M,N] = act(X)[M,K] * W^T  (W is N x K row-major)
//   XT:    X stored K-major (X[k*ldx + row]) instead of row-major
//   ACT:   x <- relu(a[k]*x + b[k])  with ab = {a[0..K-1], b[0..K-1]}
//   STATS: accumulate per-output-channel sum / sumsq into stats[0..N-1], stats[N..2N-1]
// Block: 256 thr = 8 waves; block tile = 128 rows x (NTILES*16) cols; K step 16.
// ------------------------------------------------------------------
template<int NTILES, bool XT, bool ACT, bool STATS>
__global__ __launch_bounds__(256)
void wmma_gemm(const float* __restrict__ X, int ldx,
               const float* __restrict__ Wm, int ldw, int kofs,
               const float* __restrict__ ab,
               float* __restrict__ Y,
               float* __restrict__ stats,
               int K, int N)
{
  __shared__ float As[128][18];
  __shared__ float Ws[NTILES * 16][18];
  __shared__ float lsum[NTILES * 16];
  __shared__ float lsq [NTILES * 16];

  const int tid  = threadIdx.x;
  const int lane = tid & 31;
  const int wave = tid >> 5;
  const int rowBase = blockIdx.x * 128;
  const int colBase = blockIdx.y * (NTILES * 16);

  if constexpr (STATS) {
    if (tid < NTILES * 16) { lsum[tid] = 0.0f; lsq[tid] = 0.0f; }
  }

  v8f acc[NTILES] = {};

  for (int k0 = 0; k0 < K; k0 += 16) {
    __syncthreads();
    // ---- stage A: 128 rows x 16 k ----
#pragma unroll
    for (int i = 0; i < 8; ++i) {
      int idx = tid + i * 256;
      int r, kk;
      if constexpr (XT) { kk = idx >> 7; r = idx & 127; }   // coalesced over rows
      else              { r = idx >> 4;  kk = idx & 15; }   // coalesced over k
      int gk = k0 + kk;
      float v;
      if constexpr (XT) v = X[(size_t)gk * ldx + rowBase + r];
      else              v = X[(size_t)(rowBase + r) * ldx + gk];
      if constexpr (ACT) {
        v = v * ab[gk] + ab[K + gk];
        v = v > 0.0f ? v : 0.0f;
      }
      As[r][kk] = v;
    }
    // ---- stage B: (NTILES*16) cols x 16 k, W row-major N x ldw ----
    for (int idx = tid; idx < NTILES * 16 * 16; idx += 256) {
      int n = idx >> 4, kk = idx & 15;
      Ws[n][kk] = Wm[(size_t)(colBase + n) * ldw + kofs + k0 + kk];
    }
    __syncthreads();

    // ---- compute: 4 WMMA k-steps per staged tile ----
    const int mr = wave * 16 + (lane & 15);
    const int kh = (lane >> 4) << 1;            // lanes 0-15: K {0,1}; 16-31: K {2,3}
#pragma unroll
    for (int kb = 0; kb < 16; kb += 4) {
      v2f a;
      a.x = As[mr][kb + kh];
      a.y = As[mr][kb + kh + 1];
#pragma unroll
      for (int t = 0; t < NTILES; ++t) {
        v2f b;
        b.x = Ws[t * 16 + (lane & 15)][kb + kh];
        b.y = Ws[t * 16 + (lane & 15)][kb + kh + 1];
        acc[t] = __builtin_amdgcn_wmma_f32_16x16x4_f32(
            false, a, false, b, (short)0, acc[t], false, false);
      }
    }
  }

  // ---- epilogue: store + per-channel stats ----
  const int mrow = 8 * (lane >> 4);             // C layout: VGPR j -> M = j (+8 upper half)
#pragma unroll
  for (int t = 0; t < NTILES; ++t) {
    float s = 0.0f, s2 = 0.0f;
    int nloc = t * 16 + (lane & 15);
    int n = colBase + nloc;
#pragma unroll
    for (int j = 0; j < 8; ++j) {
      float v = acc[t][j];
      int row = rowBase + wave * 16 + mrow + j;
      Y[(size_t)row * N + n] = v;
      if constexpr (STATS) { s += v; s2 += v * v; }
    }
    if constexpr (STATS) {
      atomicAdd(&lsum[nloc], s);
      atomicAdd(&lsq[nloc], s2);
    }
  }
  if constexpr (STATS) {
    __syncthreads();
    if (tid < NTILES * 16) {
      atomicAdd(&stats[colBase + tid], lsum[tid]);
      atomicAdd(&stats[N + colBase + tid], lsq[tid]);
    }
  }
}

// ------------------------------------------------------------------
// Fused conv1: y1[n*P + p][o] = t1[p][o] + bilinear(g[.,.,o], coords(p)+delta_n)
// Integer deltas => frac weights shared across n; only corner indices shift.
// g, t1 layout: (P, 96) row-major; float4 over channels. Also BN1 stats.
// grid = (P/256, 81), block = 256
// ------------------------------------------------------------------
__global__ __launch_bounds__(256)
void sample_fuse1(const float* __restrict__ coords,
                  const float* __restrict__ t1,
                  const float* __restrict__ g,
                  float* __restrict__ y1,
                  float* __restrict__ stats)
{
  __shared__ float lsum[96];
  __shared__ float lsq [96];
  const int tid = threadIdx.x;
  if (tid < 96) { lsum[tid] = 0.0f; lsq[tid] = 0.0f; }
  __syncthreads();

  const int p = blockIdx.x * 256 + tid;
  const int n = blockIdx.y;
  const int dxn = n / 9 - 4;   // delta[...,0] pairs with the slower displacement index
  const int dyn = n % 9 - 4;

  const float cx = coords[p];
  const float cy = coords[P_ + p];
  const float fx = floorf(cx), fy = floorf(cy);
  const float ax = cx - fx, ay = cy - fy;      // wx1, wy1
  const int x0 = (int)fx + dxn, y0 = (int)fy + dyn;
  const int x1 = x0 + 1,        y1c = y0 + 1;

  const bool vx0 = (x0 >= 0) & (x0 < W_);
  const bool vx1 = (x1 >= 0) & (x1 < W_);
  const bool vy0 = (y0 >= 0) & (y0 < H_);
  const bool vy1 = (y1c >= 0) & (y1c < H_);

  const float w00 = (vy0 & vx0) ? (1.0f - ay) * (1.0f - ax) : 0.0f;
  const float w01 = (vy0 & vx1) ? (1.0f - ay) * ax          : 0.0f;
  const float w10 = (vy1 & vx0) ? ay * (1.0f - ax)          : 0.0f;
  const float w11 = (vy1 & vx1) ? ay * ax                   : 0.0f;

  const int xc0 = min(max(x0, 0), W_ - 1);
  const int xc1 = min(max(x1, 0), W_ - 1);
  const int yc0 = min(max(y0, 0), H_ - 1);
  const int yc1 = min(max(y1c, 0), H_ - 1);

  const float4* g4  = (const float4*)g;
  const float4* t4  = (const float4*)t1;
  float4*       y4  = (float4*)y1;
  const size_t b00 = (size_t)(yc0 * W_ + xc0) * 24;
  const size_t b01 = (size_t)(yc0 * W_ + xc1) * 24;
  const size_t b10 = (size_t)(yc1 * W_ + xc0) * 24;
  const size_t b11 = (size_t)(yc1 * W_ + xc1) * 24;
  const size_t tb  = (size_t)p * 24;
  const size_t ob  = ((size_t)n * P_ + p) * 24;

#pragma unroll 4
  for (int o4 = 0; o4 < 24; ++o4) {
    float4 t = t4[tb + o4];
    float4 a = g4[b00 + o4], b = g4[b01 + o4], c = g4[b10 + o4], d = g4[b11 + o4];
    float4 v;
    v.x = t.x + w00 * a.x + w01 * b.x + w10 * c.x + w11 * d.x;
    v.y = t.y + w00 * a.y + w01 * b.y + w10 * c.y + w11 * d.y;
    v.z = t.z + w00 * a.z + w01 * b.z + w10 * c.z + w11 * d.z;
    v.w = t.w + w00 * a.w + w01 * b.w + w10 * c.w + w11 * d.w;
    y4[ob + o4] = v;
    int o = o4 * 4;
    atomicAdd(&lsum[o + 0], v.x); atomicAdd(&lsq[o + 0], v.x * v.x);
    atomicAdd(&lsum[o + 1], v.y); atomicAdd(&lsq[o + 1], v.y * v.y);
    atomicAdd(&lsum[o + 2], v.z); atomicAdd(&lsq[o + 2], v.z * v.z);
    atomicAdd(&lsum[o + 3], v.w); atomicAdd(&lsq[o + 3], v.w * v.w);
  }
  __syncthreads();
  if (tid < 96) {
    atomicAdd(&stats[tid],      lsum[tid]);
    atomicAdd(&stats[96 + tid], lsq[tid]);
  }
}

// ------------------------------------------------------------------
// Fold BN (train-mode batch stats) into affine a,b for the next layer's input.
// ------------------------------------------------------------------
__global__ void bn_finalize(const float* __restrict__ stats,
                            const float* __restrict__ gamma,
                            const float* __restrict__ beta,
                            float* __restrict__ ab, int N, float invM)
{
  int i = threadIdx.x;
  if (i >= N) return;
  float mean = stats[i] * invM;
  float var  = stats[N + i] * invM - mean * mean;
  float a    = gamma[i] * rsqrtf(var + EPS_);
  ab[i]     = a;
  ab[N + i] = beta[i] - mean * a;
}

// ------------------------------------------------------------------
// cost = relu(bn3(y3)) . w4 + b4 ; out = wdap . cost (over the 81 k-channels)
// block handles 64 pixels for all 81 displacements. grid = P/64
// ------------------------------------------------------------------
__global__ __launch_bounds__(256)
void cost_dap(const float* __restrict__ y3, const float* __restrict__ ab3,
              const float* __restrict__ w4, const float* __restrict__ b4,
              const float* __restrict__ wdap, float* __restrict__ out)
{
  __shared__ float costl[81][65];
  __shared__ float a3[64], b3[64], w4s[64];
  const int tid = threadIdx.x;
  if (tid < 64) { a3[tid] = ab3[tid]; b3[tid] = ab3[64 + tid]; w4s[tid] = w4[tid]; }
  __syncthreads();

  const int pBase = blockIdx.x * 64;
  const int pl = tid & 63;
  const float bias = b4[0];

  for (int k = tid >> 6; k < 81; k += 4) {
    const float* row = y3 + ((size_t)k * P_ + pBase + pl) * 64;
    float s = bias;
#pragma unroll 8
    for (int c = 0; c < 64; ++c) {
      float v = row[c] * a3[c] + b3[c];
      v = v > 0.0f ? v : 0.0f;
      s += v * w4s[c];
    }
    costl[k][pl] = s;
  }
  __syncthreads();
  for (int l = tid >> 6; l < 81; l += 4) {
    float s = 0.0f;
#pragma unroll 9
    for (int k = 0; k < 81; ++k) s += wdap[l * 81 + k] * costl[k][pl];
    out[(size_t)l * P_ + pBase + pl] = s;
  }
}

// ------------------------------------------------------------------
extern "C" void kernel_launch(void* const* d_in, const int* in_sizes, int n_in,
                              void* d_out, int out_size, void* d_ws, size_t ws_size,
                              hipStream_t stream) {
  const float* f1     = (const float*)d_in[0];   // (128, 64, 128)
  const float* f2     = (const float*)d_in[1];
  const float* coords = (const float*)d_in[2];   // (2, 64, 128)
  const float* w1     = (const float*)d_in[3];   // (96, 256)
  const float* g1     = (const float*)d_in[4];
  const float* be1    = (const float*)d_in[5];
  const float* w2     = (const float*)d_in[6];   // (128, 96)
  const float* g2     = (const float*)d_in[7];
  const float* be2    = (const float*)d_in[8];
  const float* w3     = (const float*)d_in[9];   // (64, 128)
  const float* g3     = (const float*)d_in[10];
  const float* be3    = (const float*)d_in[11];
  const float* w4     = (const float*)d_in[12];  // (1, 64)
  const float* b4     = (const float*)d_in[13];
  const float* wdap   = (const float*)d_in[14];  // (81, 81)
  float* out = (float*)d_out;

  float* ws = (float*)d_ws;
  float* stats1 = ws + OFF_STATS1;
  float* stats2 = ws + OFF_STATS2;
  float* stats3 = ws + OFF_STATS3;
  float* ab1    = ws + OFF_AB1;
  float* ab2    = ws + OFF_AB2;
  float* ab3    = ws + OFF_AB3;
  float* t1     = ws + OFF_T1;
  float* gbuf   = ws + OFF_G;
  float* y1     = ws + OFF_Y1;     // later reused as y3
  float* y2     = ws + OFF_Y2;

  // 0) zero the stats accumulators (ws is poisoned)
  init_zero<<<3, 256, 0, stream>>>(ws, 576);

  // 1) pre-GEMMs: t1 = f1^T . W1a^T, g = f2^T . W1b^T  (M=8192, K=128, N=96)
  {
    dim3 grid(P_ / 128, 1);
    wmma_gemm<6, true, false, false><<<grid, 256, 0, stream>>>(
        f1, P_, w1, 256, 0,   nullptr, t1,   nullptr, 128, 96);
    wmma_gemm<6, true, false, false><<<grid, 256, 0, stream>>>(
        f2, P_, w1, 256, 128, nullptr, gbuf, nullptr, 128, 96);
  }

  // 2) fused conv1 via shifted bilinear sampling of g; accumulate BN1 stats
  sample_fuse1<<<dim3(P_ / 256, ND_), 256, 0, stream>>>(coords, t1, gbuf, y1, stats1);
  bn_finalize<<<1, 96, 0, stream>>>(stats1, g1, be1, ab1, 96, 1.0f / (float)MTOT_);

  // 3) GEMM2: y2 = relu(bn1(y1)) . w2^T  (M=663552, K=96, N=128) + BN2 stats
  wmma_gemm<4, false, true, true><<<dim3(MTOT_ / 128, 2), 256, 0, stream>>>(
      y1, 96, w2, 96, 0, ab1, y2, stats2, 96, 128);
  bn_finalize<<<1, 128, 0, stream>>>(stats2, g2, be2, ab2, 128, 1.0f / (float)MTOT_);

  // 4) GEMM3: y3 = relu(bn2(y2)) . w3^T  (M=663552, K=128, N=64) + BN3 stats
  float* y3 = y1;  // reuse y1 region
  wmma_gemm<4, false, true, true><<<dim3(MTOT_ / 128, 1), 256, 0, stream>>>(
      y2, 128, w3, 128, 0, ab2, y3, stats3, 128, 64);
  bn_finalize<<<1, 64, 0, stream>>>(stats3, g3, be3, ab3, 64, 1.0f / (float)MTOT_);

  // 5) fused final conv (64->1, +bias) and 81x81 DAP mix
  cost_dap<<<P_ / 64, 256, 0, stream>>>(y3, ab3, w4, b4, wdap, out);
}